// MaskedGat_contrastive_linegraphs_32753420599851
// MI455X (gfx1250) — compile-verified
//
#include <hip/hip_runtime.h>
#include <hip/hip_bf16.h>

#define NN 50000
#define EE 800000
#define IND 128
#define NH  4
#define DD  32
#define HD  128
#define CC  16

typedef float v2f __attribute__((ext_vector_type(2)));
typedef float v8f __attribute__((ext_vector_type(8)));

__device__ __forceinline__ unsigned fenc(float f) {
    unsigned u = __float_as_uint(f);
    return (u & 0x80000000u) ? ~u : (u | 0x80000000u);
}
__device__ __forceinline__ float fdec(unsigned u) {
    u = (u & 0x80000000u) ? (u & 0x7fffffffu) : ~u;
    return __uint_as_float(u);
}
__device__ __forceinline__ float lrelu(float v) { return v > 0.f ? v : 0.2f * v; }

// ---------------- utility ----------------
__global__ void zero_kernel(float* p, int n) {
    int i = blockIdx.x * blockDim.x + threadIdx.x;
    if (i < n) p[i] = 0.0f;
}

// h0 = mask ? mask_token : x
__global__ void mask_kernel(const float* __restrict__ x, const float* __restrict__ tok,
                            const unsigned char* __restrict__ mask, float* __restrict__ h) {
    int i = blockIdx.x * blockDim.x + threadIdx.x;
    if (i >= NN * IND) return;
    int node = i >> 7, c = i & 127;
    h[i] = mask[node] ? tok[c] : x[i];
}

// ---------------- feat = h @ W (N x 128 x 128), fp32 WMMA ----------------
// LDS holds W pair-interleaved: element (k,n) at dword (k>>1)*256 + 2n + (k&1),
// so a WMMA B-fragment (K, K+1 for one n) is a single contiguous 8B DS load.
// All 8 B fragments of a K-chunk are loaded into distinct registers before the
// WMMAs so the DS loads can clause and their waits stagger across the MACs.
__global__ void feat_gemm(const float* __restrict__ h, const float* __restrict__ W,
                          float* __restrict__ feat, int nrows) {
    __shared__ float sW2[IND * HD]; // 64 KB
    int tid = threadIdx.x;
    for (int i = tid; i < IND * HD; i += 128) {
        int k = i >> 7, n = i & 127;
        sW2[((k >> 1) << 8) + (n << 1) + (k & 1)] = W[i];
    }
    __syncthreads();

    int wave = tid >> 5;
    int lane = tid & 31;
    int m0 = (blockIdx.x * 4 + wave) * 16;
    if (m0 >= nrows) return;          // wave-uniform: EXEC stays all-ones for WMMA

    int row   = m0 + (lane & 15);
    int khalf = (lane >> 4) << 1;     // 0 or 2
    int ncol  = lane & 15;

    v8f acc[8] = {};
    const float* hrow = h + row * IND;
    for (int kc = 0; kc < IND; kc += 4) {
        v2f a;
        a.x = hrow[kc + khalf];
        a.y = hrow[kc + khalf + 1];
        const float* bp = &sW2[(((kc + khalf) >> 1) << 8) + (ncol << 1)];
        v2f b0 = *(const v2f*)(bp + 0 * 32);
        v2f b1 = *(const v2f*)(bp + 1 * 32);
        v2f b2 = *(const v2f*)(bp + 2 * 32);
        v2f b3 = *(const v2f*)(bp + 3 * 32);
        v2f b4 = *(const v2f*)(bp + 4 * 32);
        v2f b5 = *(const v2f*)(bp + 5 * 32);
        v2f b6 = *(const v2f*)(bp + 6 * 32);
        v2f b7 = *(const v2f*)(bp + 7 * 32);
        acc[0] = __builtin_amdgcn_wmma_f32_16x16x4_f32(false, a, false, b0, (short)0, acc[0], false, false);
        acc[1] = __builtin_amdgcn_wmma_f32_16x16x4_f32(false, a, false, b1, (short)0, acc[1], false, false);
        acc[2] = __builtin_amdgcn_wmma_f32_16x16x4_f32(false, a, false, b2, (short)0, acc[2], false, false);
        acc[3] = __builtin_amdgcn_wmma_f32_16x16x4_f32(false, a, false, b3, (short)0, acc[3], false, false);
        acc[4] = __builtin_amdgcn_wmma_f32_16x16x4_f32(false, a, false, b4, (short)0, acc[4], false, false);
        acc[5] = __builtin_amdgcn_wmma_f32_16x16x4_f32(false, a, false, b5, (short)0, acc[5], false, false);
        acc[6] = __builtin_amdgcn_wmma_f32_16x16x4_f32(false, a, false, b6, (short)0, acc[6], false, false);
        acc[7] = __builtin_amdgcn_wmma_f32_16x16x4_f32(false, a, false, b7, (short)0, acc[7], false, false);
    }
    // C/D layout: vgpr j -> row m0 + j + 8*(lane/16), col t*16 + lane%16
    int srow = m0 + ((lane >> 4) << 3);
#pragma unroll
    for (int t = 0; t < 8; ++t)
#pragma unroll
        for (int j = 0; j < 8; ++j)
            feat[(srow + j) * HD + t * 16 + ncol] = acc[t][j];
}

// el/er: per (node, head) dot of feat[n, head*32 : +32] with attn vectors
__global__ void eler_kernel(const float* __restrict__ feat, const float* __restrict__ al,
                            const float* __restrict__ ar, float* __restrict__ el,
                            float* __restrict__ er) {
    int i = blockIdx.x * blockDim.x + threadIdx.x;
    if (i >= NN * NH) return;
    int node = i >> 2, head = i & 3;
    const float* f = feat + node * HD + head * DD;
    const float* a = al + head * DD;
    const float* b = ar + head * DD;
    float sl = 0.f, sr = 0.f;
#pragma unroll 8
    for (int d = 0; d < DD; ++d) { float v = f[d]; sl += v * a[d]; sr += v * b[d]; }
    el[i] = sl; er[i] = sr;
}

// per edge: e = leaky_relu(el[src]+er[dst]) for 4 heads; segment max via ordered-uint atomicMax
__global__ void edge_e_kernel(const int* __restrict__ src, const int* __restrict__ dst,
                              const float* __restrict__ el, const float* __restrict__ er,
                              float* __restrict__ ebuf, unsigned* __restrict__ menc) {
    int e = blockIdx.x * blockDim.x + threadIdx.x;
    if (e >= EE) return;
    int s = src[e], d = dst[e];
    const float4 l4 = *(const float4*)(el + s * NH);
    const float4 r4 = *(const float4*)(er + d * NH);
    float4 v;
    v.x = lrelu(l4.x + r4.x);
    v.y = lrelu(l4.y + r4.y);
    v.z = lrelu(l4.z + r4.z);
    v.w = lrelu(l4.w + r4.w);
    *(float4*)(ebuf + e * NH) = v;
    unsigned* m = menc + d * NH;
    atomicMax(m + 0, fenc(v.x));
    atomicMax(m + 1, fenc(v.y));
    atomicMax(m + 2, fenc(v.z));
    atomicMax(m + 3, fenc(v.w));
}

// per edge: ex = exp(e - m[dst]); segment sum for 4 heads
__global__ void edge_exp_kernel(const int* __restrict__ dst, float* __restrict__ ebuf,
                                const unsigned* __restrict__ menc, float* __restrict__ ssum) {
    int e = blockIdx.x * blockDim.x + threadIdx.x;
    if (e >= EE) return;
    int d = dst[e];
    const uint4 m4 = *(const uint4*)(menc + d * NH);
    float4 v = *(float4*)(ebuf + e * NH);
    v.x = expf(v.x - fdec(m4.x));
    v.y = expf(v.y - fdec(m4.y));
    v.z = expf(v.z - fdec(m4.z));
    v.w = expf(v.w - fdec(m4.w));
    *(float4*)(ebuf + e * NH) = v;
    float* s = ssum + d * NH;
    atomicAdd(s + 0, v.x);
    atomicAdd(s + 1, v.y);
    atomicAdd(s + 2, v.z);
    atomicAdd(s + 3, v.w);
}

// ssum -> 1 / max(ssum, 1e-9): turns per-edge-lane division into a multiply
__global__ void inv_kernel(float* __restrict__ ssum, int n) {
    int i = blockIdx.x * blockDim.x + threadIdx.x;
    if (i < n) ssum[i] = 1.0f / fmaxf(ssum[i], 1e-9f);
}

// rst[dst] += feat[src] * alpha   (32 threads per edge, float4 per thread)
__global__ void edge_msg_kernel(const int* __restrict__ src, const int* __restrict__ dst,
                                const float* __restrict__ ebuf, const float* __restrict__ sinv,
                                const float* __restrict__ feat, float* __restrict__ rst) {
    long long i = (long long)blockIdx.x * blockDim.x + threadIdx.x;
    if (i >= (long long)EE * 32) return;
    int e  = (int)(i >> 5);
    int t  = (int)(i & 31);
    int head = t >> 3;          // 4 heads
    int q    = t & 7;           // 8 float4 chunks per head
    int s = src[e], d = dst[e];
    float alpha = ebuf[e * NH + head] * sinv[d * NH + head];
    const float4 f = *(const float4*)(feat + s * HD + head * DD + q * 4);
    float* r = rst + d * HD + head * DD + q * 4;
    atomicAdd(r + 0, f.x * alpha);
    atomicAdd(r + 1, f.y * alpha);
    atomicAdd(r + 2, f.z * alpha);
    atomicAdd(r + 3, f.w * alpha);
}

// h = relu(rst + h + bias)
__global__ void node_update_kernel(float* __restrict__ h, const float* __restrict__ rst,
                                   const float* __restrict__ bias) {
    int i = blockIdx.x * blockDim.x + threadIdx.x;
    if (i >= NN * HD) return;
    float v = rst[i] + h[i] + bias[i & 127];
    h[i] = v > 0.f ? v : 0.f;
}

// per-feature sum / sumsq (block = 128 threads = one thread per feature, 256-row slab)
__global__ void bn_stats_kernel(const float* __restrict__ h, float* __restrict__ bsum,
                                float* __restrict__ bsq) {
    int c  = threadIdx.x;
    int r0 = blockIdx.x * 256;
    int r1 = r0 + 256; if (r1 > NN) r1 = NN;
    float s = 0.f, s2 = 0.f;
    for (int r = r0; r < r1; ++r) { float v = h[r * HD + c]; s += v; s2 += v * v; }
    atomicAdd(&bsum[c], s);
    atomicAdd(&bsq[c], s2);
}

// fold BN into FC: Weff[k][c] = a[k]*fcW[k][c]; beff[c] = fcb[c] + sum_k shift[k]*fcW[k][c]
__global__ void bn_prep_kernel(const float* __restrict__ bsum, const float* __restrict__ bsq,
                               const float* __restrict__ gamma, const float* __restrict__ beta,
                               const float* __restrict__ fcW, const float* __restrict__ fcb,
                               float* __restrict__ Weff, float* __restrict__ beff) {
    __shared__ float sh[HD];
    int c = threadIdx.x; // 128
    float mu  = bsum[c] * (1.0f / NN);
    float var = bsq[c] * (1.0f / NN) - mu * mu;
    float a   = rsqrtf(var + 1e-5f) * gamma[c];
    sh[c] = beta[c] - mu * a;
#pragma unroll
    for (int j = 0; j < CC; ++j) Weff[c * CC + j] = a * fcW[c * CC + j];
    __syncthreads();
    if (c < CC) {
        float b = fcb[c];
        for (int k = 0; k < HD; ++k) b += sh[k] * fcW[k * CC + c];
        beff[c] = b;
    }
}

// out = h @ Weff + beff (N x 128 x 16), fp32 WMMA; wave: 16 rows x 16 cols
__global__ void fc_gemm(const float* __restrict__ h, const float* __restrict__ Weff,
                        const float* __restrict__ beff, float* __restrict__ out, int nrows) {
    __shared__ float sW2[IND * CC]; // 8 KB, pair-interleaved
    int tid = threadIdx.x;
    for (int i = tid; i < IND * CC; i += 128) {
        int k = i >> 4, n = i & 15;
        sW2[((k >> 1) << 5) + (n << 1) + (k & 1)] = Weff[i];
    }
    __syncthreads();

    int wave = tid >> 5;
    int lane = tid & 31;
    int m0 = (blockIdx.x * 4 + wave) * 16;
    if (m0 >= nrows) return;

    int row   = m0 + (lane & 15);
    int khalf = (lane >> 4) << 1;
    int ncol  = lane & 15;

    v8f acc = {};
    const float* hrow = h + row * IND;
    for (int kc = 0; kc < IND; kc += 4) {
        v2f a;
        a.x = hrow[kc + khalf];
        a.y = hrow[kc + khalf + 1];
        v2f b = *(const v2f*)&sW2[(((kc + khalf) >> 1) << 5) + (ncol << 1)];
        acc = __builtin_amdgcn_wmma_f32_16x16x4_f32(
            false, a, false, b, (short)0, acc, false, false);
    }
    float bb = beff[ncol];
    int srow = m0 + ((lane >> 4) << 3);
#pragma unroll
    for (int j = 0; j < 8; ++j)
        out[(srow + j) * CC + ncol] = acc[j] + bb;
}

extern "C" void kernel_launch(void* const* d_in, const int* in_sizes, int n_in,
                              void* d_out, int out_size, void* d_ws, size_t ws_size,
                              hipStream_t stream) {
    const float* x      = (const float*)d_in[0];
    const float* tok    = (const float*)d_in[1];
    const float* W      = (const float*)d_in[2];   // [2,128,128]
    const float* al     = (const float*)d_in[3];   // [2,4,32]
    const float* ar     = (const float*)d_in[4];
    const float* bias   = (const float*)d_in[5];   // [2,128]
    const float* gamma  = (const float*)d_in[6];
    const float* beta   = (const float*)d_in[7];
    const float* fcW    = (const float*)d_in[8];   // [128,16]
    const float* fcb    = (const float*)d_in[9];
    const int*   srcp   = (const int*)d_in[10];
    const int*   dstp   = (const int*)d_in[11];
    const unsigned char* maskp = (const unsigned char*)d_in[12];

    float* ws   = (float*)d_ws;
    float* h    = ws;                         // N*128
    float* feat = ws + (size_t)NN * HD;       // N*128
    float* rst  = feat + (size_t)NN * HD;     // N*128
    float* ebuf = rst + (size_t)NN * HD;      // E*4
    float* el   = ebuf + (size_t)EE * NH;     // N*4
    float* er   = el + (size_t)NN * NH;       // N*4
    unsigned* menc = (unsigned*)(er + (size_t)NN * NH); // N*4
    float* ssum = (float*)(menc + (size_t)NN * NH);     // N*4
    float* bsum = ssum + (size_t)NN * NH;     // 128
    float* bsq  = bsum + HD;                  // 128
    float* Weff = bsq + HD;                   // 2048
    float* beff = Weff + HD * CC;             // 16

    const int gemm_blocks = (NN / 16 + 3) / 4; // 782 (N % 16 == 0)

    mask_kernel<<<(NN * IND + 255) / 256, 256, 0, stream>>>(x, tok, maskp, h);

    for (int l = 0; l < 2; ++l) {
        feat_gemm<<<gemm_blocks, 128, 0, stream>>>(h, W + (size_t)l * IND * HD, feat, NN);
        eler_kernel<<<(NN * NH + 255) / 256, 256, 0, stream>>>(
            feat, al + l * NH * DD, ar + l * NH * DD, el, er);
        zero_kernel<<<(NN * HD + 255) / 256, 256, 0, stream>>>(rst, NN * HD);
        zero_kernel<<<(NN * NH + 255) / 256, 256, 0, stream>>>((float*)menc, NN * NH);
        zero_kernel<<<(NN * NH + 255) / 256, 256, 0, stream>>>(ssum, NN * NH);
        edge_e_kernel<<<(EE + 255) / 256, 256, 0, stream>>>(srcp, dstp, el, er, ebuf, menc);
        edge_exp_kernel<<<(EE + 255) / 256, 256, 0, stream>>>(dstp, ebuf, menc, ssum);
        inv_kernel<<<(NN * NH + 255) / 256, 256, 0, stream>>>(ssum, NN * NH);
        edge_msg_kernel<<<(int)(((long long)EE * 32 + 255) / 256), 256, 0, stream>>>(
            srcp, dstp, ebuf, ssum, feat, rst);
        node_update_kernel<<<(NN * HD + 255) / 256, 256, 0, stream>>>(h, rst, bias + l * HD);
    }

    zero_kernel<<<1, 256, 0, stream>>>(bsum, 2 * HD); // bsum + bsq contiguous
    bn_stats_kernel<<<(NN + 255) / 256, 128, 0, stream>>>(h, bsum, bsq);
    bn_prep_kernel<<<1, 128, 0, stream>>>(bsum, bsq, gamma, beta, fcW, fcb, Weff, beff);
    fc_gemm<<<gemm_blocks, 128, 0, stream>>>(h, Weff, beff, (float*)d_out, NN);
}